// GRIT_encoder_last_layer_67886253081132
// MI455X (gfx1250) — compile-verified
//
#include <hip/hip_runtime.h>
#include <math.h>

// ---------------- problem constants (from setup_inputs) ----------------
#define BB   4
#define NN   512
#define CC   128
#define ED   64
#define ED2  128
#define ATT  8
#define INCH 128

typedef float v2f __attribute__((ext_vector_type(2)));
typedef float v8f __attribute__((ext_vector_type(8)));

__device__ __forceinline__ float gelu_exact(float x) {
    return 0.5f * x * (1.0f + erff(x * 0.70710678118654752f));
}

// ---------------- tiny prologue kernels ----------------
// q[b][o] = node[b,0,:] . Wq_w[o,:] + Wq_b[o]   (o in 0..1023)
__global__ void k_qvec(const float* __restrict__ node, const float* __restrict__ Wq_w,
                       const float* __restrict__ Wq_b, float* __restrict__ q) {
    int b = blockIdx.x;
    int o = blockIdx.y * blockDim.x + threadIdx.x;       // 0..1023
    const float* nrow = node + b * NN * CC;              // node[b,0,:]
    float s = Wq_b[o];
    const float* wr = Wq_w + o * CC;
    for (int c = 0; c < CC; ++c) s += nrow[c] * wr[c];
    q[b * (ED2 * ATT) + o] = s;
}

// Weff[b][e][c] = sum_a q[b][a*128+e] * Wk_w[a*128+e][c] ; beff similarly with Wk_b
__global__ void k_weff(const float* __restrict__ q, const float* __restrict__ Wk_w,
                       const float* __restrict__ Wk_b, float* __restrict__ Weff,
                       float* __restrict__ beff) {
    int b = blockIdx.x;
    int e = blockIdx.y;        // 0..127
    int c = threadIdx.x;       // 0..127
    float s = 0.0f;
    for (int a = 0; a < ATT; ++a) {
        float qv = q[b * (ED2 * ATT) + a * ED2 + e];
        s += qv * Wk_w[(a * ED2 + e) * CC + c];
    }
    Weff[(b * ED2 + e) * CC + c] = s;
    if (c == 0) {
        float t = 0.0f;
        for (int a = 0; a < ATT; ++a)
            t += q[b * (ED2 * ATT) + a * ED2 + e] * Wk_b[a * ED2 + e];
        beff[b * ED2 + e] = t;
    }
}

// ---------------- generic wave-level f32 WMMA GEMM ----------------
// D[b][m][n] = act( sum_k A[b][m][k] * W[b][n][k] + bias[b][n] )
// One wave (32 threads) computes one 16x16 tile using V_WMMA_F32_16X16X4_F32.
__global__ void k_wmma_gemm(const float* __restrict__ A, int ars, int astride_b,
                            const float* __restrict__ W, int wstride_b,
                            const float* __restrict__ bias, int bstride_b,
                            float* __restrict__ D, int drs, int dstride_b,
                            int K, int act) {
    const int lane = threadIdx.x;          // 0..31, all active (EXEC all-1s)
    const int row0 = blockIdx.x * 16;
    const int col0 = blockIdx.y * 16;
    const int b    = blockIdx.z;

    const float* Ab = A + (size_t)b * astride_b;
    const float* Wb = W + (size_t)b * wstride_b;
    float*       Db = D + (size_t)b * dstride_b;

    const int mi = row0 + (lane & 15);     // A row this lane feeds
    const int ni = col0 + (lane & 15);     // B column this lane feeds
    const int ks = (lane >> 4) * 2;        // lanes 0-15: K offsets {0,1}; 16-31: {2,3}

    const float* arow = Ab + (size_t)mi * ars + ks;
    const float* wrow = Wb + (size_t)ni * K + ks;

    v8f acc = {};
    for (int k = 0; k < K; k += 4) {
        v2f a, w;
        a.x = arow[k];     a.y = arow[k + 1];
        w.x = wrow[k];     w.y = wrow[k + 1];
        // D = A(16x4) x B(4x16) + C   (fp32 WMMA)
        acc = __builtin_amdgcn_wmma_f32_16x16x4_f32(
                  false, a, false, w, (short)0, acc, false, false);
    }

    const float bv = bias ? bias[b * bstride_b + ni] : 0.0f;
    const int mbase = row0 + 8 * (lane >> 4);   // C/D layout: M = r + 8*(lane>=16)
    for (int r = 0; r < 8; ++r) {
        float v = acc[r] + bv;
        if (act) v = gelu_exact(v);
        Db[(size_t)(mbase + r) * drs + ni] = v;
    }
}

// ---------------- elementwise h = gelu(rho(qk1*ew) + eb + qk2 + emb) ----------------
__global__ void k_h(const float* __restrict__ qk, const float* __restrict__ ew,
                    const float* __restrict__ eb, const float* __restrict__ emb,
                    float* __restrict__ h) {
    int idx = blockIdx.x * blockDim.x + threadIdx.x;   // B*N*ED = 131072
    int j = idx & (ED - 1);
    int n = (idx >> 6) & (NN - 1);
    int b = idx >> 15;
    size_t qbase = (size_t)(b * NN + n) * ED2;
    float x1 = qk[qbase + j];
    float x2 = qk[qbase + ED + j];
    size_t ebase = (size_t)(b * NN + n) * ED + j;
    float ewv = ew[ebase];
    float ebv = eb[ebase];
    float em  = emb[(size_t)b * NN * NN * ED + (size_t)n * ED + j]; // embedding[b,0,n,j]
    float t = x1 * ewv;
    float rho = copysignf(sqrtf(fabsf(t)), t);
    h[ebase] = gelu_exact(rho + ebv + x2 + em);
}

// ---------------- softmax over n = 1..511 for each (b, l) column ----------------
__global__ void k_softmax(const float* __restrict__ s, float* __restrict__ alphas) {
    int b = blockIdx.x, l = blockIdx.y;
    int tid = threadIdx.x;                 // 128 threads
    __shared__ float red[128];

    float m = -INFINITY;
    for (int n = 1 + tid; n < NN; n += 128)
        m = fmaxf(m, s[(size_t)(b * NN + n) * INCH + l]);
    red[tid] = m; __syncthreads();
    for (int o = 64; o > 0; o >>= 1) { if (tid < o) red[tid] = fmaxf(red[tid], red[tid + o]); __syncthreads(); }
    m = red[0]; __syncthreads();

    float sum = 0.0f;
    for (int n = 1 + tid; n < NN; n += 128)
        sum += expf(s[(size_t)(b * NN + n) * INCH + l] - m);
    red[tid] = sum; __syncthreads();
    for (int o = 64; o > 0; o >>= 1) { if (tid < o) red[tid] += red[tid + o]; __syncthreads(); }
    sum = red[0];

    float inv = 1.0f / sum;
    for (int n = 1 + tid; n < NN; n += 128) {
        size_t i = (size_t)(b * NN + n) * INCH + l;
        alphas[i] = expf(s[i] - m) * inv;
    }
}

// ---------------- build concat([et,nt]) (zero-padded row 511) + emit et_p ----------------
__global__ void k_scale(const float* __restrict__ etlin, const float* __restrict__ ntlin,
                        const float* __restrict__ alphas, float* __restrict__ cat,
                        float* __restrict__ out_et_p) {
    int idx = blockIdx.x * blockDim.x + threadIdx.x;   // B*512*128
    int l = idx & (INCH - 1);
    int m = (idx >> 7) & (NN - 1);
    int b = idx >> 16;
    size_t crow = (size_t)(b * NN + m) * (2 * INCH);
    if (m == NN - 1) {                                  // pad row: zeros
        cat[crow + l] = 0.0f;
        cat[crow + INCH + l] = 0.0f;
        return;
    }
    size_t i = (size_t)(b * NN + m + 1) * INCH + l;     // node n = m+1
    float a  = alphas[i];
    float ev = etlin[i] * a;
    float nv = ntlin[i] * a;
    cat[crow + l]        = ev;
    cat[crow + INCH + l] = nv;
    // et_p shape (B, 128, 1, 511): idx = b*128*511 + l*511 + m
    out_et_p[(size_t)b * INCH * (NN - 1) + (size_t)l * (NN - 1) + m] = ev;
}

// ---------------- final: node_out = scaler * sum_m tmp ; emit tmp_p ----------------
__global__ void k_final(const float* __restrict__ tmp, float* __restrict__ node_out,
                        float* __restrict__ out_tmp_p) {
    int b = blockIdx.x, l = blockIdx.y;
    int tid = threadIdx.x;                 // 128 threads
    __shared__ float red[128];
    float s = 0.0f;
    for (int m = tid; m < NN - 1; m += 128) {
        float v = tmp[(size_t)(b * NN + m) * INCH + l];
        out_tmp_p[(size_t)b * INCH * (NN - 1) + (size_t)l * (NN - 1) + m] = v;
        s += v;
    }
    red[tid] = s; __syncthreads();
    for (int o = 64; o > 0; o >>= 1) { if (tid < o) red[tid] += red[tid + o]; __syncthreads(); }
    if (tid == 0)
        node_out[b * INCH + l] = red[0] * 0.17677669529663689f;   // 2/sqrt(128)
}

// ---------------- host side ----------------
extern "C" void kernel_launch(void* const* d_in, const int* in_sizes, int n_in,
                              void* d_out, int out_size, void* d_ws, size_t ws_size,
                              hipStream_t stream) {
    const float* node = (const float*)d_in[0];
    const float* edge = (const float*)d_in[1];
    const float* emb  = (const float*)d_in[2];
    const float* Wq_w = (const float*)d_in[3];
    const float* Wq_b = (const float*)d_in[4];
    const float* Wk_w = (const float*)d_in[5];
    const float* Wk_b = (const float*)d_in[6];
    const float* Wew  = (const float*)d_in[7];
    const float* Web  = (const float*)d_in[8];
    const float* Wa   = (const float*)d_in[9];
    const float* nt_w = (const float*)d_in[10];
    const float* nt_b = (const float*)d_in[11];
    const float* et_w = (const float*)d_in[12];
    const float* et_b = (const float*)d_in[13];
    const float* h1_w = (const float*)d_in[14];
    const float* h1_b = (const float*)d_in[15];
    const float* h2_w = (const float*)d_in[16];
    const float* h2_b = (const float*)d_in[17];

    float* out      = (float*)d_out;
    float* node_out = out;                                 // 4*128
    float* tmp_p    = out + BB * INCH;                     // 4*128*511
    float* et_p     = tmp_p + BB * INCH * (NN - 1);        // 4*128*511

    // workspace layout (floats)
    float* ws     = (float*)d_ws;
    float* q      = ws;                  // 4096
    float* Weff   = q + 4096;            // 65536
    float* beff   = Weff + 65536;        // 512
    float* qk     = beff + 512;          // 262144
    float* ew     = qk + 262144;         // 131072
    float* eb     = ew + 131072;         // 131072
    float* h      = eb + 131072;         // 131072
    float* s      = h + 131072;          // 262144
    float* alphas = s + 262144;          // 262144
    float* etlin  = alphas + 262144;     // 262144
    float* ntlin  = etlin + 262144;      // 262144
    float* cat    = ntlin + 262144;      // 524288
    float* g      = cat + 524288;        // 262144
    float* tmpb   = g + 262144;          // 262144

    // 1) q projection (only node row 0 per batch)
    k_qvec<<<dim3(BB, 4), 256, 0, stream>>>(node, Wq_w, Wq_b, q);
    // 2) fold heads: effective per-batch k-projection weight/bias
    k_weff<<<dim3(BB, ED2), CC, 0, stream>>>(q, Wk_w, Wk_b, Weff, beff);
    // 3) qk[b,n,e] = node[b,n,:] . Weff[b,e,:] + beff   (M=512,N=128,K=128 per batch)
    k_wmma_gemm<<<dim3(NN/16, ED2/16, BB), 32, 0, stream>>>(
        node, CC, NN*CC, Weff, ED2*CC, beff, ED2, qk, ED2, NN*ED2, CC, 0);
    // 4) ew = edge[:,0] @ Wew.T   (M=512,N=64,K=64 per batch)
    k_wmma_gemm<<<dim3(NN/16, ED/16, BB), 32, 0, stream>>>(
        edge, ED, NN*NN*ED, Wew, 0, nullptr, 0, ew, ED, NN*ED, ED, 0);
    // 5) eb = edge[:,0] @ Web.T
    k_wmma_gemm<<<dim3(NN/16, ED/16, BB), 32, 0, stream>>>(
        edge, ED, NN*NN*ED, Web, 0, nullptr, 0, eb, ED, NN*ED, ED, 0);
    // 6) h = gelu(rho(qk1*ew) + eb + qk2 + emb)
    k_h<<<(BB*NN*ED)/256, 256, 0, stream>>>(qk, ew, eb, emb, h);
    // 7) s = h @ Wa.T   (M=512,N=128,K=64)
    k_wmma_gemm<<<dim3(NN/16, INCH/16, BB), 32, 0, stream>>>(
        h, ED, NN*ED, Wa, 0, nullptr, 0, s, INCH, NN*INCH, ED, 0);
    // 8) etlin = h @ et_w.T + et_b
    k_wmma_gemm<<<dim3(NN/16, INCH/16, BB), 32, 0, stream>>>(
        h, ED, NN*ED, et_w, 0, et_b, 0, etlin, INCH, NN*INCH, ED, 0);
    // 9) ntlin = node @ nt_w.T + nt_b
    k_wmma_gemm<<<dim3(NN/16, INCH/16, BB), 32, 0, stream>>>(
        node, CC, NN*CC, nt_w, 0, nt_b, 0, ntlin, INCH, NN*INCH, CC, 0);
    // 10) softmax over nodes 1..511
    k_softmax<<<dim3(BB, INCH), 128, 0, stream>>>(s, alphas);
    // 11) build concat([et, nt]) with shift/scale; emit et_p output
    k_scale<<<(BB*NN*INCH)/256, 256, 0, stream>>>(etlin, ntlin, alphas, cat, et_p);
    // 12) g = gelu(cat @ h1_w.T + h1_b)   (M=512,N=128,K=256)
    k_wmma_gemm<<<dim3(NN/16, INCH/16, BB), 32, 0, stream>>>(
        cat, 2*INCH, NN*2*INCH, h1_w, 0, h1_b, 0, g, INCH, NN*INCH, 2*INCH, 1);
    // 13) tmp = g @ h2_w.T + h2_b   (M=512,N=128,K=128)
    k_wmma_gemm<<<dim3(NN/16, INCH/16, BB), 32, 0, stream>>>(
        g, INCH, NN*INCH, h2_w, 0, h2_b, 0, tmpb, INCH, NN*INCH, INCH, 0);
    // 14) node_out + tmp_p
    k_final<<<dim3(BB, INCH), 128, 0, stream>>>(tmpb, node_out, tmp_p);
}